// Part_Graph_55765855371288
// MI455X (gfx1250) — compile-verified
//
#include <hip/hip_runtime.h>
#include <hip/hip_bf16.h>
#include <math.h>

// ---------------------------------------------------------------------------
// Part-graph forward for MI455X (gfx1250, wave32).
//  * img conv (256x264 GEMM, 5.8 GMAC) algebraically folded into a 10x264
//    effective weight "Weff" -> proj1 is a tiny K=264 WMMA GEMM.
//  * 3600x3600 attention fused flash-style (online softmax), never
//    materialized (saves ~1.2 GB of HBM traffic; workspace ~27MB fits in L2).
//  * fp32 WMMA (V_WMMA_F32_16X16X4_F32) keeps full precision.
//  * proj1 K-loop split: branch-free main loop over p_fea rows (0..255),
//    peeled coord epilogue (rows 256..263) -> tight 2xload+ds+wmma loop body.
// ---------------------------------------------------------------------------

typedef __attribute__((ext_vector_type(2))) float v2f;
typedef __attribute__((ext_vector_type(8))) float v8f;

#define HW   3600
#define WIDTH 60
#define NB   4
#define HID  10
#define CIN  256
#define OUTC 168
#define OUTNS (OUTC*HW)   // 604800

__device__ __forceinline__ float coordf(int f, int s) {
    int x = s % WIDTH, y = s / WIDTH;
    const float inv = 1.0f / 60.0f;
    float xmin = x * inv * 2.f - 1.f;
    float ymin = y * inv * 2.f - 1.f;
    float xmax = (x + 1) * inv * 2.f - 1.f;
    float ymax = (y + 1) * inv * 2.f - 1.f;
    switch (f) {
        case 0: return xmin;
        case 1: return ymin;
        case 2: return xmax;
        case 3: return ymax;
        case 4: return 0.5f * (xmin + xmax);
        case 5: return 0.5f * (ymin + ymax);
        default: return inv;   // 6,7: 1/W, 1/H
    }
}

__device__ __forceinline__ float sigmoidf_(float x) { return 1.f / (1.f + __expf(-x)); }

// --------------------------------------------------------------------------
// Weff[k,cc] = sum_c W[c,k]*img_w[c,cc]  (k<10, zero-padded to 16 rows)
// bk[k]     = sum_c W[c,k]*img_b[c]
// --------------------------------------------------------------------------
__global__ void wk_weff(const float* __restrict__ W, const float* __restrict__ imgw,
                        const float* __restrict__ imgb,
                        float* __restrict__ weff, float* __restrict__ bk) {
    int idx = blockIdx.x * 256 + threadIdx.x;
    if (idx >= 16 * 264) return;
    int k = idx / 264, cc = idx % 264;
    float acc = 0.f;
    if (k < HID)
        for (int c = 0; c < CIN; ++c) acc += W[c * HID + k] * imgw[c * 264 + cc];
    weff[k * 264 + cc] = acc;
    if (cc == 0) {
        float bb = 0.f;
        if (k < HID)
            for (int c = 0; c < CIN; ++c) bb += W[c * HID + k] * imgb[c];
        bk[k] = bb;
    }
}

// --------------------------------------------------------------------------
// decomp attention map: m = att_w @ x + att_b ; map->d_out ; att = softmax_ch(m)
// --------------------------------------------------------------------------
__global__ void wk_small_att(const float* __restrict__ xh, const float* __restrict__ aw,
                             const float* __restrict__ ab, int C,
                             float* __restrict__ att, float* __restrict__ mapo) {
    int idx = blockIdx.x * 256 + threadIdx.x;
    if (idx >= NB * HW) return;
    int s = idx % HW, n = idx / HW;
    const float* X = xh + n * HID * HW;
    float xv[HID];
    for (int k = 0; k < HID; ++k) xv[k] = X[k * HW + s];
    float mv[5]; float mx = -3.0e38f;
    for (int c = 0; c < C; ++c) {
        float a = ab[c];
        for (int k = 0; k < HID; ++k) a += aw[c * HID + k] * xv[k];
        mv[c] = a;
        mx = fmaxf(mx, a);
        mapo[n * OUTNS + c * HW + s] = a;   // pre-softmax map (pu_map/pl_map)
    }
    float sum = 0.f;
    for (int c = 0; c < C; ++c) { float e = __expf(mv[c] - mx); mv[c] = e; sum += e; }
    float inv = 1.f / sum;
    for (int c = 0; c < C; ++c) att[(n * C + c) * HW + s] = mv[c] * inv;
}

// --------------------------------------------------------------------------
// zcat[0:10] = A * S(channel ssel)   (S==null -> 1) ; zcat[10:20] = B
// --------------------------------------------------------------------------
__global__ void wk_mul_concat(const float* __restrict__ A, const float* __restrict__ S,
                              int sct, int ssel, const float* __restrict__ B,
                              float* __restrict__ out) {
    int idx = blockIdx.x * 256 + threadIdx.x;
    if (idx >= NB * HID * HW) return;
    int s = idx % HW; int t = idx / HW; int c = t % HID; int n = t / HID;
    float sv = S ? S[(n * sct + ssel) * HW + s] : 1.f;
    out[(n * 20 + c) * HW + s]      = A[(n * HID + c) * HW + s] * sv;
    out[(n * 20 + 10 + c) * HW + s] = B[(n * HID + c) * HW + s];
}

// --------------------------------------------------------------------------
// generic 3x3 SAME conv; mode 0: +bias ; mode 1: bn(g,b)+relu
// --------------------------------------------------------------------------
__global__ void wk_conv3x3(const float* __restrict__ in, const float* __restrict__ w,
                           const float* __restrict__ g, const float* __restrict__ b,
                           float* __restrict__ out, int Cin, int Cout,
                           int out_nstride, int mode) {
    int idx = blockIdx.x * 256 + threadIdx.x;
    if (idx >= NB * Cout * HW) return;
    int s = idx % HW; int t = idx / HW; int co = t % Cout; int n = t / Cout;
    int x = s % WIDTH, y = s / WIDTH;
    const float* ip = in + n * Cin * HW;
    const float* wp = w + co * Cin * 9;
    float acc = 0.f;
    for (int ci = 0; ci < Cin; ++ci) {
        const float* irow = ip + ci * HW;
        const float* wr = wp + ci * 9;
        #pragma unroll
        for (int ky = -1; ky <= 1; ++ky) {
            int yy = y + ky; if (yy < 0 || yy >= WIDTH) continue;
            #pragma unroll
            for (int kx = -1; kx <= 1; ++kx) {
                int xx = x + kx; if (xx < 0 || xx >= WIDTH) continue;
                acc += wr[(ky + 1) * 3 + (kx + 1)] * irow[yy * WIDTH + xx];
            }
        }
    }
    if (mode == 1) acc = fmaxf(acc * g[co] + b[co], 0.f);
    else           acc += b[co];
    out[n * out_nstride + co * HW + s] = acc;
}

// --------------------------------------------------------------------------
// generic 1x1 conv; mode 1: bn(g,b)+relu
// --------------------------------------------------------------------------
__global__ void wk_conv1x1(const float* __restrict__ in, const float* __restrict__ w,
                           const float* __restrict__ g, const float* __restrict__ b,
                           float* __restrict__ out, int Cin, int Cout,
                           int out_nstride, int mode) {
    int idx = blockIdx.x * 256 + threadIdx.x;
    if (idx >= NB * Cout * HW) return;
    int s = idx % HW; int t = idx / HW; int co = t % Cout; int n = t / Cout;
    const float* ip = in + n * Cin * HW;
    float acc = 0.f;
    for (int ci = 0; ci < Cin; ++ci) acc += w[co * Cin + ci] * ip[ci * HW + s];
    if (mode == 1) acc = fmaxf(acc * g[co] + b[co], 0.f);
    else           acc += b[co];
    out[n * out_nstride + co * HW + s] = acc;
}

// --------------------------------------------------------------------------
// node features: node[k,s] = node_w @ concat(xp, coord) + node_b
// stored as [n][16][HW], rows 10..15 = 0 (zero padding for WMMA)
// --------------------------------------------------------------------------
__global__ void wk_node(const float* __restrict__ xp, const float* __restrict__ nw,
                        const float* __restrict__ nb, float* __restrict__ nodeo) {
    int idx = blockIdx.x * 256 + threadIdx.x;
    if (idx >= NB * HW) return;
    int s = idx % HW, n = idx / HW;
    const float* X = xp + n * HID * HW;
    float* O = nodeo + n * 16 * HW;
    float cf[8];
    #pragma unroll
    for (int f = 0; f < 8; ++f) cf[f] = coordf(f, s);
    for (int k = 0; k < HID; ++k) {
        float acc = nb[k];
        for (int c = 0; c < HID; ++c) acc += nw[k * 18 + c] * X[c * HW + s];
        #pragma unroll
        for (int c = 0; c < 8; ++c) acc += nw[k * 18 + 10 + c] * cf[c];
        O[k * HW + s] = acc;
    }
    for (int k = HID; k < 16; ++k) O[k * HW + s] = 0.f;
}

// --------------------------------------------------------------------------
// proj1 (WMMA GEMM):  proj1[s, k] = sum_cc Weff[k,cc] * Xc[cc,s] + bk[k]
// Xc rows 0..255 = p_fea (branch-free main loop); rows 256..263 = coord
// (peeled epilogue, values precomputed + selected by uniform half-wave flag).
// D tile layout: M=s (16), N=k (16).  One wave per (s-tile, n).
// --------------------------------------------------------------------------
__global__ void wk_proj1(const float* __restrict__ pfea, const float* __restrict__ weff,
                         const float* __restrict__ bk, float* __restrict__ proj1) {
    __shared__ float sW[16 * 264];
    int lane = threadIdx.x;
    for (int t = lane; t < 16 * 264; t += 32) sW[t] = weff[t];
    __syncthreads();

    int st = blockIdx.x, n = blockIdx.y;
    int s0 = st * 16;
    int m = lane & 15;          // s-row for A, k-col for B/D
    bool lo = lane < 16;
    const float* X = pfea + n * CIN * HW + s0 + m;   // lane's column
    const float* Wrow = sW + m * 264;

    v8f C = {};
    // ---- main loop: p_fea rows, branch free --------------------------------
    int ccsel = lo ? 0 : 2;
    for (int kb = 0; kb < CIN; kb += 4) {
        int cc = kb + ccsel;
        v2f a, b;
        a.x = X[(size_t)cc * HW];
        a.y = X[(size_t)(cc + 1) * HW];
        b.x = Wrow[cc];
        b.y = Wrow[cc + 1];
        C = __builtin_amdgcn_wmma_f32_16x16x4_f32(false, a, false, b,
                                                  (short)0, C, false, false);
    }
    // ---- coord epilogue: rows 256..263 ------------------------------------
    {
        int s = s0 + m;
        float c0 = coordf(0, s), c1 = coordf(1, s), c2 = coordf(2, s), c3 = coordf(3, s);
        float c4 = coordf(4, s), c5 = coordf(5, s), c6 = coordf(6, s), c7 = coordf(7, s);
        v2f a, b;
        // kb = 256: rows 256..259 -> coord 0..3
        a.x = lo ? c0 : c2;  a.y = lo ? c1 : c3;
        b.x = Wrow[256 + ccsel]; b.y = Wrow[256 + ccsel + 1];
        C = __builtin_amdgcn_wmma_f32_16x16x4_f32(false, a, false, b,
                                                  (short)0, C, false, false);
        // kb = 260: rows 260..263 -> coord 4..7
        a.x = lo ? c4 : c6;  a.y = lo ? c5 : c7;
        b.x = Wrow[260 + ccsel]; b.y = Wrow[260 + ccsel + 1];
        C = __builtin_amdgcn_wmma_f32_16x16x4_f32(false, a, false, b,
                                                  (short)0, C, false, false);
    }
    float bias = bk[m];
    float* P = proj1 + n * HW * 16;
    #pragma unroll
    for (int r = 0; r < 8; ++r) {
        int s = s0 + r + (lo ? 0 : 8);
        P[s * 16 + m] = C[r] + bias;
    }
}

// --------------------------------------------------------------------------
// Fused attention (flash style), one wave per (q-tile, n):
//   scores S[s,q] = sum_k node[k,s] * proj1[q,k]        (WMMA, M=s N=q)
//   online softmax over s (per-lane 8 + shfl_xor(16) partner combine)
//   O[c,q] += sum_s xp[c,s] * P[s,q]                    (WMMA, M=c N=q)
// --------------------------------------------------------------------------
__global__ void wk_attn(const float* __restrict__ proj1, const float* __restrict__ nodeb,
                        const float* __restrict__ xp, float* __restrict__ co) {
    int lane = threadIdx.x;
    int qt = blockIdx.x, n = blockIdx.y;
    int q0 = qt * 16;
    int m = lane & 15;
    bool lo = lane < 16;

    const float* P1 = proj1 + n * HW * 16;
    const float* ND = nodeb + n * 16 * HW;
    const float* V  = xp + n * HID * HW;

    // Q in B-layout, reused for all 225 s-tiles
    v2f bq[4];
    #pragma unroll
    for (int j = 0; j < 4; ++j) {
        int cc = 4 * j + (lo ? 0 : 2);
        bq[j].x = P1[(q0 + m) * 16 + cc];
        bq[j].y = P1[(q0 + m) * 16 + cc + 1];
    }
    int cv = (m < HID) ? m : (HID - 1);  // clamp padded c rows (never stored)
    const float* Vrow = V + cv * HW;

    v8f O = {};
    float m_run = -3.0e38f, l_run = 0.f;

    for (int st = 0; st < 225; ++st) {
        int s0 = st * 16;
        // ---- scores tile
        v8f S = {};
        #pragma unroll
        for (int j = 0; j < 4; ++j) {
            int kk = 4 * j + (lo ? 0 : 2);
            v2f a;
            a.x = ND[kk * HW + s0 + m];
            a.y = ND[(kk + 1) * HW + s0 + m];
            S = __builtin_amdgcn_wmma_f32_16x16x4_f32(false, a, false, bq[j],
                                                      (short)0, S, false, false);
        }
        // ---- online softmax over s (rows); lane L & L+16 share query column
        float mx = S[0];
        #pragma unroll
        for (int r = 1; r < 8; ++r) mx = fmaxf(mx, S[r]);
        mx = fmaxf(mx, __shfl_xor(mx, 16, 32));
        float m_new = fmaxf(m_run, mx);
        float scale = __expf(m_run - m_new);
        float p[8]; float ls = 0.f;
        #pragma unroll
        for (int r = 0; r < 8; ++r) { p[r] = __expf(S[r] - m_new); ls += p[r]; }
        ls += __shfl_xor(ls, 16, 32);
        l_run = l_run * scale + ls;
        m_run = m_new;
        #pragma unroll
        for (int r = 0; r < 8; ++r) O[r] *= scale;
        // ---- O += V * P   (rebuild P in B-layout via shuffles)
        #pragma unroll
        for (int j = 0; j < 4; ++j) {
            int kb = 4 * j;
            int srcl = m + ((kb < 8) ? 0 : 16);
            int rb = kb & 4;
            float t0 = __shfl(p[rb + 0], srcl, 32);
            float t1 = __shfl(p[rb + 1], srcl, 32);
            float t2 = __shfl(p[rb + 2], srcl, 32);
            float t3 = __shfl(p[rb + 3], srcl, 32);
            v2f bp; bp.x = lo ? t0 : t2; bp.y = lo ? t1 : t3;
            int srel = kb + (lo ? 0 : 2);
            v2f av; av.x = Vrow[s0 + srel]; av.y = Vrow[s0 + srel + 1];
            O = __builtin_amdgcn_wmma_f32_16x16x4_f32(false, av, false, bp,
                                                      (short)0, O, false, false);
        }
    }
    float inv = 1.f / l_run;
    float* Co = co + n * HID * HW;
    #pragma unroll
    for (int r = 0; r < 8; ++r) {
        int c = r + (lo ? 0 : 8);
        if (c < HID) Co[c * HW + q0 + m] = O[r] * inv;
    }
}

// --------------------------------------------------------------------------
// atts[i] = softmax_ch( att_w @ (dp_w @ p_fea + dp_b) + att_b )
// --------------------------------------------------------------------------
__global__ void wk_atts(const float* __restrict__ pfea, const float* __restrict__ dpw,
                        const float* __restrict__ dpb, const float* __restrict__ aw,
                        const float* __restrict__ ab, int C, float* __restrict__ out) {
    int idx = blockIdx.x * 256 + threadIdx.x;
    if (idx >= NB * HW) return;
    int s = idx % HW, n = idx / HW;
    const float* F = pfea + n * CIN * HW;
    float t[HID];
    for (int k = 0; k < HID; ++k) t[k] = dpb[k];
    for (int c = 0; c < CIN; ++c) {
        float f = F[c * HW + s];
        for (int k = 0; k < HID; ++k) t[k] += dpw[k * CIN + c] * f;
    }
    float av[3]; float mx = -3.0e38f;
    for (int c = 0; c < C; ++c) {
        float a = ab[c];
        for (int k = 0; k < HID; ++k) a += aw[c * HID + k] * t[k];
        av[c] = a; mx = fmaxf(mx, a);
    }
    float sum = 0.f;
    for (int c = 0; c < C; ++c) { float e = __expf(av[c] - mx); av[c] = e; sum += e; }
    float inv = 1.f / sum;
    for (int c = 0; c < C; ++c) out[(n * C + c) * HW + s] = av[c] * inv;
}

// --------------------------------------------------------------------------
// GRU helper kernels
// --------------------------------------------------------------------------
__global__ void wk_gru_mix(const float* __restrict__ msg, const float* __restrict__ gates,
                           const float* __restrict__ h, float* __restrict__ out) {
    int idx = blockIdx.x * 256 + threadIdx.x;
    if (idx >= NB * HID * HW) return;
    int s = idx % HW; int t = idx / HW; int c = t % HID; int n = t / HID;
    float r = sigmoidf_(gates[(n * 20 + c) * HW + s]);
    out[(n * 20 + c) * HW + s]      = msg[(n * HID + c) * HW + s];
    out[(n * 20 + 10 + c) * HW + s] = r * h[(n * HID + c) * HW + s];
}

__global__ void wk_gru_out(const float* __restrict__ gates, const float* __restrict__ cand,
                           const float* __restrict__ h, float* __restrict__ out) {
    int idx = blockIdx.x * 256 + threadIdx.x;
    if (idx >= NB * HID * HW) return;
    int s = idx % HW; int t = idx / HW; int c = t % HID; int n = t / HID;
    float zg = sigmoidf_(gates[(n * 20 + 10 + c) * HW + s]);
    float cd = tanhf(cand[(n * HID + c) * HW + s]);
    float hv = h[(n * HID + c) * HW + s];
    out[n * OUTNS + c * HW + s] = (1.f - zg) * hv + zg * cd;
}

// ===========================================================================
// Host launcher
// ===========================================================================
extern "C" void kernel_launch(void* const* d_in, const int* in_sizes, int n_in,
                              void* d_out, int out_size, void* d_ws, size_t ws_size,
                              hipStream_t stream) {
    auto IN = [&](int i) { return (const float*)d_in[i]; };

    // pytree flatten order (sorted dict keys):
    // 0:xf 1:xh 2:xp_parts 3:p_fea
    // ctx(4..27, stride4: att_b,att_w,dp_b,dp_w)
    // dec_l(28..35: att_b,att_w,fh_b1,fh_b2,fh_g1,fh_g2,fh_w1,fh_w2)
    // dec_u(36..43: same)
    // fcont(44..91, stride8: W,img_b,img_w,node_b,node_w,proj_b,proj_g,proj_w)
    // gru(92..115, stride4: bc,bg,wc,wg)
    // pdep(116..175, stride6: b1,b2,g1,g2,w1,w2)
    const int CX = 4, DL = 28, DU = 36, FC = 44, GR = 92, PD = 116;

    const float* xh0  = IN(1);
    const float* xh1  = IN(1) + NB * HID * HW;
    const float* pfea = IN(3);
    auto xp = [&](int i) { return IN(2) + i * NB * HID * HW; };

    float* ws = (float*)d_ws;
    float* dout = (float*)d_out;
    // workspace layout (floats)
    const size_t OFF_ATTU = 0;                       // [4,5,HW]
    const size_t OFF_ATTL = OFF_ATTU + 72000;        // [4,3,HW]
    const size_t OFF_ZA   = OFF_ATTL + 43200;        // [4,20,HW]
    const size_t OFF_ZB   = OFF_ZA + 288000;         // [4,20,HW]
    const size_t OFF_ZC   = OFF_ZB + 288000;         // [4,20,HW] gates
    const size_t OFF_CAND = OFF_ZC + 288000;         // [4,10,HW]
    const size_t OFF_PU   = OFF_CAND + 144000;       // 6 x [4,10,HW] (pu0-3, pl0-1)
    const size_t OFF_WEFF = OFF_PU + 6 * 144000;     // 6 x [16,264]
    const size_t OFF_BK   = OFF_WEFF + 6 * 4224;     // 6 x [16]
    const size_t OFF_PROJ = OFF_BK + 96;             // 6 x [4,HW,16]
    const size_t OFF_NODE = OFF_PROJ + 6 * 230400;   // 6 x [4,16,HW]
    const size_t OFF_CO   = OFF_NODE + 6 * 230400;   // 6 x [4,10,HW]
    const size_t OFF_FDEP = OFF_CO + 6 * 144000;     // 6 x [4,10,HW]
    const size_t OFF_ATTS = OFF_FDEP + 6 * 144000;   // 6 x [4,<=3,HW]

    auto nb = [](int t) { return (t + 255) / 256; };
    const int T_NS  = NB * HW;            // 14400
    const int T_NCS = NB * HID * HW;      // 144000

    // ---- Weff per part ----------------------------------------------------
    for (int i = 0; i < 6; ++i)
        wk_weff<<<nb(16 * 264), 256, 0, stream>>>(
            IN(FC + 8 * i + 0), IN(FC + 8 * i + 2), IN(FC + 8 * i + 1),
            ws + OFF_WEFF + i * 4224, ws + OFF_BK + i * 16);

    // ---- decomp upper (4 parts) & lower (2 parts) -------------------------
    wk_small_att<<<nb(T_NS), 256, 0, stream>>>(xh0, IN(DU + 1), IN(DU + 0), 5,
                                               ws + OFF_ATTU, dout + 160 * HW);
    wk_small_att<<<nb(T_NS), 256, 0, stream>>>(xh1, IN(DL + 1), IN(DL + 0), 3,
                                               ws + OFF_ATTL, dout + 165 * HW);
    for (int p = 0; p < 4; ++p) {
        wk_mul_concat<<<nb(T_NCS), 256, 0, stream>>>(xh0, ws + OFF_ATTU, 5, p + 1,
                                                     xp(p), ws + OFF_ZA);
        wk_conv3x3<<<nb(NB * 20 * HW), 256, 0, stream>>>(
            ws + OFF_ZA, IN(DU + 6), IN(DU + 4), IN(DU + 2), ws + OFF_ZB, 20, 20, 20 * HW, 1);
        wk_conv3x3<<<nb(NB * 10 * HW), 256, 0, stream>>>(
            ws + OFF_ZB, IN(DU + 7), IN(DU + 5), IN(DU + 3),
            ws + OFF_PU + p * 144000, 20, 10, 10 * HW, 1);
    }
    for (int p = 0; p < 2; ++p) {
        wk_mul_concat<<<nb(T_NCS), 256, 0, stream>>>(xh1, ws + OFF_ATTL, 3, p + 1,
                                                     xp(4 + p), ws + OFF_ZA);
        wk_conv3x3<<<nb(NB * 20 * HW), 256, 0, stream>>>(
            ws + OFF_ZA, IN(DL + 6), IN(DL + 4), IN(DL + 2), ws + OFF_ZB, 20, 20, 20 * HW, 1);
        wk_conv3x3<<<nb(NB * 10 * HW), 256, 0, stream>>>(
            ws + OFF_ZB, IN(DL + 7), IN(DL + 5), IN(DL + 3),
            ws + OFF_PU + (4 + p) * 144000, 20, 10, 10 * HW, 1);
    }

    // ---- context per part: node, proj1(WMMA), attention(WMMA), F_dep, atts
    const int PCNT[6] = {2, 3, 3, 3, 3, 2};   // atts channel count per part
    for (int i = 0; i < 6; ++i) {
        wk_node<<<nb(T_NS), 256, 0, stream>>>(xp(i), IN(FC + 8 * i + 4),
                                              IN(FC + 8 * i + 3),
                                              ws + OFF_NODE + i * 230400);
        wk_proj1<<<dim3(225, NB), 32, 0, stream>>>(
            pfea, ws + OFF_WEFF + i * 4224, ws + OFF_BK + i * 16,
            ws + OFF_PROJ + i * 230400);
        wk_attn<<<dim3(225, NB), 32, 0, stream>>>(
            ws + OFF_PROJ + i * 230400, ws + OFF_NODE + i * 230400, xp(i),
            ws + OFF_CO + i * 144000);
        wk_conv1x1<<<nb(NB * 10 * HW), 256, 0, stream>>>(
            ws + OFF_CO + i * 144000, IN(FC + 8 * i + 7), IN(FC + 8 * i + 6),
            IN(FC + 8 * i + 5), ws + OFF_FDEP + i * 144000, 10, 10, 10 * HW, 1);
        wk_atts<<<nb(T_NS), 256, 0, stream>>>(
            pfea, IN(CX + 4 * i + 3), IN(CX + 4 * i + 2), IN(CX + 4 * i + 1),
            IN(CX + 4 * i + 0), PCNT[i], ws + OFF_ATTS + i * 43200);
    }

    // ---- per-edge Part_Dependency ----------------------------------------
    const int EU[10] = {0, 1, 1, 2, 2, 3, 3, 4, 4, 5};
    const int EV[10] = {1, 0, 2, 1, 3, 2, 4, 3, 5, 4};
    const int EC[10] = {1, 1, 2, 1, 2, 1, 2, 1, 2, 1};  // 1 + PART_LIST[u].index(v)
    for (int e = 0; e < 10; ++e) {
        int u = EU[e], v = EV[e];
        wk_mul_concat<<<nb(T_NCS), 256, 0, stream>>>(
            ws + OFF_FDEP + u * 144000, ws + OFF_ATTS + u * 43200, PCNT[u], EC[e],
            xp(v), ws + OFF_ZA);
        wk_conv3x3<<<nb(NB * 20 * HW), 256, 0, stream>>>(
            ws + OFF_ZA, IN(PD + 6 * e + 4), IN(PD + 6 * e + 2), IN(PD + 6 * e + 0),
            ws + OFF_ZB, 20, 20, 20 * HW, 1);
        wk_conv1x1<<<nb(NB * 10 * HW), 256, 0, stream>>>(
            ws + OFF_ZB, IN(PD + 6 * e + 5), IN(PD + 6 * e + 3), IN(PD + 6 * e + 1),
            dout + (60 + 10 * e) * HW, 20, 10, OUTNS, 1);
    }

    // ---- ConvGRU node updates --------------------------------------------
    for (int i = 0; i < 6; ++i) {
        const float* msg = ws + OFF_PU + i * 144000;
        wk_mul_concat<<<nb(T_NCS), 256, 0, stream>>>(msg, nullptr, 0, 0, xp(i),
                                                     ws + OFF_ZA);
        wk_conv3x3<<<nb(NB * 20 * HW), 256, 0, stream>>>(
            ws + OFF_ZA, IN(GR + 4 * i + 3), nullptr, IN(GR + 4 * i + 1),
            ws + OFF_ZC, 20, 20, 20 * HW, 0);                       // gates = conv+bg
        wk_gru_mix<<<nb(T_NCS), 256, 0, stream>>>(msg, ws + OFF_ZC, xp(i), ws + OFF_ZA);
        wk_conv3x3<<<nb(NB * 10 * HW), 256, 0, stream>>>(
            ws + OFF_ZA, IN(GR + 4 * i + 2), nullptr, IN(GR + 4 * i + 0),
            ws + OFF_CAND, 20, 10, 10 * HW, 0);                     // cand_raw = conv+bc
        wk_gru_out<<<nb(T_NCS), 256, 0, stream>>>(ws + OFF_ZC, ws + OFF_CAND, xp(i),
                                                  dout + (10 * i) * HW);
    }

    (void)in_sizes; (void)n_in; (void)out_size; (void)ws_size;
}